// NestedGCN_41661182771861
// MI455X (gfx1250) — compile-verified
//
#include <hip/hip_runtime.h>
#include <math.h>

#define N_NODES 50000
#define N_EDGES 800000
#define D_FEAT  128
#define HIDDEN  128
#define N_LAYERS 4
#define N_SUB   5000
#define N_GRAPH 128
#define N_CLASS 10
#define FEAT_DIM (N_LAYERS*HIDDEN)   /* 512 */

typedef __attribute__((ext_vector_type(2))) float v2f;
typedef __attribute__((ext_vector_type(8))) float v8f;

// ---------------- workspace layout (float offsets) ----------------
// all float4-accessed bases are 16B aligned
static const size_t OFF_DINV = 0;                                   // 50000
static const size_t OFF_M    = OFF_DINV + N_NODES;                  // 6,400,000
static const size_t OFF_AGG  = OFF_M    + (size_t)N_NODES*HIDDEN;   // 6,400,000
static const size_t OFF_FEAT = OFF_AGG  + (size_t)N_NODES*HIDDEN;   // 25,600,000
static const size_t OFF_SUB  = OFF_FEAT + (size_t)N_NODES*FEAT_DIM; // 2,560,000
static const size_t OFF_SUBC = OFF_SUB  + (size_t)N_SUB*FEAT_DIM;   // 5,000
static const size_t OFF_G    = OFF_SUBC + N_SUB;                    // 65,536
static const size_t OFF_GC   = OFF_G    + (size_t)N_GRAPH*FEAT_DIM; // 128
static const size_t OFF_HID  = OFF_GC   + N_GRAPH;                  // 16,384

// ---------------- small utility kernels ----------------
__global__ void zero_f32_k(float* __restrict__ p, int n) {
    int i = blockIdx.x * blockDim.x + threadIdx.x;
    if (i < n) p[i] = 0.0f;
}

__global__ void init_deg_k(float* __restrict__ deg) {
    int i = blockIdx.x * blockDim.x + threadIdx.x;
    if (i < N_NODES) deg[i] = 1.0f;   // self loop
}

__global__ void count_deg_k(const int* __restrict__ dst, float* __restrict__ deg) {
    int e = blockIdx.x * blockDim.x + threadIdx.x;
    if (e < N_EDGES) atomicAdd(&deg[dst[e]], 1.0f);
}

__global__ void finalize_dinv_k(float* __restrict__ deg) {
    int i = blockIdx.x * blockDim.x + threadIdx.x;
    if (i < N_NODES) deg[i] = rsqrtf(deg[i]);   // deg >= 1 always
}

// ---------------- WMMA GEMM: out[M,128] = A[M,K] (lda) @ W[K,128] ----------------
// mode 0: out = m, agg = m * dinv[row]^2        (seed self-loop term)
// mode 1: out = relu(m + bias)                  (final MLP layer 1)
__global__ void __launch_bounds__(256) gemm_wmma_k(
    const float* __restrict__ A, int lda, int K,
    const float* __restrict__ W,
    const float* __restrict__ bias,
    const float* __restrict__ dinv,
    float* __restrict__ out,
    float* __restrict__ agg,
    int mode)
{
    __shared__ float As[16 * 516];               // supports K up to 512, +4 pad/row
    const int row0 = blockIdx.x * 16;
    const int ldas = K + 4;                      // bank-conflict-free column reads
    for (int idx = threadIdx.x; idx < 16 * K; idx += 256) {
        int r = idx / K;
        int c = idx - r * K;
        As[r * ldas + c] = A[(size_t)(row0 + r) * lda + c];
    }
    __syncthreads();

    const int lane = threadIdx.x & 31;
    const int wave = threadIdx.x >> 5;           // 8 waves -> 8 column tiles
    const int half = lane >> 4;                  // K-pair select (ISA 16x4 f32 layout)
    const int m16  = lane & 15;                  // A: M row, B: N col
    const int col0 = wave * 16;

    v8f acc = {};
    for (int k0 = 0; k0 < K; k0 += 4) {
        const int k = k0 + 2 * half;
        v2f a, b;
        a.x = As[m16 * ldas + k];
        a.y = As[m16 * ldas + k + 1];
        b.x = W[(size_t)k       * HIDDEN + col0 + m16];
        b.y = W[(size_t)(k + 1) * HIDDEN + col0 + m16];
        acc = __builtin_amdgcn_wmma_f32_16x16x4_f32(
            false, a, false, b, (short)0, acc, false, false);
    }

#pragma unroll
    for (int v = 0; v < 8; ++v) {
        int r = row0 + v + 8 * half;             // C/D layout: VGPR v -> M = v (+8 hi half)
        int c = col0 + m16;
        float val = acc[v];
        if (mode == 0) {
            out[(size_t)r * HIDDEN + c] = val;
            float dv = dinv[r];
            agg[(size_t)r * HIDDEN + c] = val * dv * dv;
        } else {
            float t = val + bias[c];
            out[(size_t)r * HIDDEN + c] = t > 0.0f ? t : 0.0f;
        }
    }
}

// ---------------- edge scatter: agg[dst] += m[src] * dinv[src]*dinv[dst] ----------------
__global__ void scatter_edges_k(const int* __restrict__ src, const int* __restrict__ dst,
                                const float* __restrict__ dinv, const float* __restrict__ m,
                                float* __restrict__ agg)
{
    int tid = blockIdx.x * blockDim.x + threadIdx.x;    // N_EDGES * 32 threads
    if (tid >= N_EDGES * 32) return;
    int e  = tid >> 5;
    int c4 = tid & 31;
    int s = src[e], d = dst[e];
    float w = dinv[s] * dinv[d];
    const float4 v = *reinterpret_cast<const float4*>(m + (size_t)s * HIDDEN + c4 * 4);
    float* b = agg + (size_t)d * HIDDEN + c4 * 4;
    atomicAdd(b + 0, v.x * w);
    atomicAdd(b + 1, v.y * w);
    atomicAdd(b + 2, v.z * w);
    atomicAdd(b + 3, v.w * w);
}

// ---------------- bias + relu into concatenated feat column block ----------------
__global__ void bias_relu_k(const float* __restrict__ agg, const float* __restrict__ b,
                            float* __restrict__ feat, int layer)
{
    int tid = blockIdx.x * blockDim.x + threadIdx.x;    // N_NODES * 128
    if (tid >= N_NODES * HIDDEN) return;
    int node = tid >> 7;
    int c    = tid & 127;
    float t = agg[(size_t)node * HIDDEN + c] + b[c];
    feat[(size_t)node * FEAT_DIM + layer * HIDDEN + c] = t > 0.0f ? t : 0.0f;
}

// ---------------- segment mean accumulators ----------------
__global__ void sub_accum_k(const float* __restrict__ feat, const int* __restrict__ n2s,
                            float* __restrict__ sum, float* __restrict__ cnt)
{
    int tid = blockIdx.x * blockDim.x + threadIdx.x;    // N_NODES * 128 (float4 chunks)
    if (tid >= N_NODES * 128) return;
    int node = tid >> 7;
    int c4   = tid & 127;
    int sg = n2s[node];
    const float4 v = *reinterpret_cast<const float4*>(feat + (size_t)node * FEAT_DIM + c4 * 4);
    float* b = sum + (size_t)sg * FEAT_DIM + c4 * 4;
    atomicAdd(b + 0, v.x); atomicAdd(b + 1, v.y);
    atomicAdd(b + 2, v.z); atomicAdd(b + 3, v.w);
    if (c4 == 0) atomicAdd(&cnt[sg], 1.0f);
}

__global__ void g_accum_k(const float* __restrict__ sub, const int* __restrict__ s2g,
                          float* __restrict__ sum, float* __restrict__ cnt)
{
    int tid = blockIdx.x * blockDim.x + threadIdx.x;    // N_SUB * 128
    if (tid >= N_SUB * 128) return;
    int s  = tid >> 7;
    int c4 = tid & 127;
    int g = s2g[s];
    const float4 v = *reinterpret_cast<const float4*>(sub + (size_t)s * FEAT_DIM + c4 * 4);
    float* b = sum + (size_t)g * FEAT_DIM + c4 * 4;
    atomicAdd(b + 0, v.x); atomicAdd(b + 1, v.y);
    atomicAdd(b + 2, v.z); atomicAdd(b + 3, v.w);
    if (c4 == 0) atomicAdd(&cnt[g], 1.0f);
}

__global__ void seg_div_k(float* __restrict__ p, const float* __restrict__ cnt, int n)
{
    int i = blockIdx.x * blockDim.x + threadIdx.x;      // n = n_seg * 512
    if (i < n) p[i] /= fmaxf(cnt[i >> 9], 1.0f);
}

// ---------------- final: logits + log_softmax ----------------
__global__ void final_lsm_k(const float* __restrict__ hidden,
                            const float* __restrict__ W2, const float* __restrict__ b2,
                            float* __restrict__ out)
{
    int r = threadIdx.x;                                 // 128 graphs, one block
    if (r >= N_GRAPH) return;
    float logit[N_CLASS];
    for (int c = 0; c < N_CLASS; ++c) {
        float acc = b2[c];
        for (int k = 0; k < HIDDEN; ++k)
            acc += hidden[(size_t)r * HIDDEN + k] * W2[(size_t)k * N_CLASS + c];
        logit[c] = acc;
    }
    float mx = logit[0];
    for (int c = 1; c < N_CLASS; ++c) mx = fmaxf(mx, logit[c]);
    float s = 0.0f;
    for (int c = 0; c < N_CLASS; ++c) s += expf(logit[c] - mx);
    float lse = mx + logf(s);
    for (int c = 0; c < N_CLASS; ++c) out[(size_t)r * N_CLASS + c] = logit[c] - lse;
}

// ---------------- host ----------------
extern "C" void kernel_launch(void* const* d_in, const int* in_sizes, int n_in,
                              void* d_out, int out_size, void* d_ws, size_t ws_size,
                              hipStream_t stream)
{
    const float* x     = (const float*)d_in[0];
    const int*   edge  = (const int*)d_in[1];
    const int*   src   = edge;
    const int*   dst   = edge + N_EDGES;
    const int*   n2s   = (const int*)d_in[2];
    const int*   s2g   = (const int*)d_in[3];
    const float* W1    = (const float*)d_in[4];
    const float* b1    = (const float*)d_in[5];
    const float* convW = (const float*)d_in[6];
    const float* convb = (const float*)d_in[7];
    const float* lin1W = (const float*)d_in[8];
    const float* lin1b = (const float*)d_in[9];
    const float* lin2W = (const float*)d_in[10];
    const float* lin2b = (const float*)d_in[11];

    float* ws   = (float*)d_ws;
    float* dinv = ws + OFF_DINV;
    float* m    = ws + OFF_M;
    float* agg  = ws + OFF_AGG;
    float* feat = ws + OFF_FEAT;
    float* ssum = ws + OFF_SUB;
    float* scnt = ws + OFF_SUBC;
    float* gsum = ws + OFF_G;
    float* gcnt = ws + OFF_GC;
    float* hid  = ws + OFF_HID;

    const int B = 256;

    // degree / norm (self loop included -> deg >= 1)
    init_deg_k<<<(N_NODES + B - 1) / B, B, 0, stream>>>(dinv);
    count_deg_k<<<(N_EDGES + B - 1) / B, B, 0, stream>>>(dst, dinv);
    finalize_dinv_k<<<(N_NODES + B - 1) / B, B, 0, stream>>>(dinv);

    // 4 GCN layers
    for (int l = 0; l < N_LAYERS; ++l) {
        const float* Ain = (l == 0) ? x : (feat + (size_t)(l - 1) * HIDDEN);
        int          lda = (l == 0) ? D_FEAT : FEAT_DIM;
        const float* Wl  = (l == 0) ? W1 : (convW + (size_t)(l - 1) * HIDDEN * HIDDEN);
        const float* bl  = (l == 0) ? b1 : (convb + (size_t)(l - 1) * HIDDEN);

        gemm_wmma_k<<<N_NODES / 16, B, 0, stream>>>(Ain, lda, HIDDEN, Wl,
                                                    nullptr, dinv, m, agg, 0);
        scatter_edges_k<<<(N_EDGES * 32 + B - 1) / B, B, 0, stream>>>(src, dst, dinv, m, agg);
        bias_relu_k<<<(N_NODES * HIDDEN + B - 1) / B, B, 0, stream>>>(agg, bl, feat, l);
    }

    // zero all segment accumulators (contiguous region)
    {
        int nz = (int)(N_SUB * FEAT_DIM + N_SUB + N_GRAPH * FEAT_DIM + N_GRAPH);
        zero_f32_k<<<(nz + B - 1) / B, B, 0, stream>>>(ssum, nz);
    }

    // node -> subgraph mean
    sub_accum_k<<<(N_NODES * 128 + B - 1) / B, B, 0, stream>>>(feat, n2s, ssum, scnt);
    seg_div_k<<<(N_SUB * FEAT_DIM + B - 1) / B, B, 0, stream>>>(ssum, scnt, N_SUB * FEAT_DIM);

    // subgraph -> graph mean
    g_accum_k<<<(N_SUB * 128 + B - 1) / B, B, 0, stream>>>(ssum, s2g, gsum, gcnt);
    seg_div_k<<<(N_GRAPH * FEAT_DIM + B - 1) / B, B, 0, stream>>>(gsum, gcnt, N_GRAPH * FEAT_DIM);

    // hidden = relu(g @ lin1_W + lin1_b)  -- same WMMA kernel, K=512, epilogue mode 1
    gemm_wmma_k<<<N_GRAPH / 16, B, 0, stream>>>(gsum, FEAT_DIM, FEAT_DIM, lin1W,
                                                lin1b, nullptr, hid, nullptr, 1);

    // logits + log_softmax
    final_lsm_k<<<1, 128, 0, stream>>>(hid, lin2W, lin2b, (float*)d_out);
}